// MultiResolutionHashEncoding_7327214207034
// MI455X (gfx1250) — compile-verified
//
#include <hip/hip_runtime.h>
#include <hip/hip_bf16.h>

// InstantNGP-style multiresolution hash encoding for MI455X (gfx1250).
// L2-gather-bound: 64 MB table set stays L2-resident (RT policy on gathers);
// x (12 MB) and out (128 MB) are streamed with non-temporal hints.
// Fully unrolled over levels: all scalings are compile-time power-of-two
// constants (exact, bit-identical to the reference), result row is held in
// registers and written as 8 back-to-back NT b128 stores so each 128 B
// output line is fully combined before NT stream-out.

#define L_LEVELS 16
#define F_FEAT   2
#define LOG2T    19
#define T_SIZE   (1u << LOG2T)

typedef float v4f __attribute__((ext_vector_type(4)));

struct alignas(8) F2 { float a, b; };

__device__ __forceinline__ void
level_interp(const F2* __restrict__ tbl,
             float px, float py, float pz, float pxp, float pyp, float pzp,
             float sc, float gr, float& ra, float& rb)
{
    const uint32_t P1  = 2654435761u;
    const uint32_t P2  = 805459861u;
    const uint32_t msk = T_SIZE - 1u;

    // bl = floor((p+1)/grid); 1/grid == sc exactly (power of two).
    const float fx = floorf(pxp * sc);
    const float fy = floorf(pyp * sc);
    const float fz = floorf(pzp * sc);

    // w = (p - vmin)/grid, vmin = f*gr - 1; all scalings exact.
    const float wx = (px - (fx * gr - 1.0f)) * sc;
    const float wy = (py - (fy * gr - 1.0f)) * sc;
    const float wz = (pz - (fz * gr - 1.0f)) * sc;
    const float ix = 1.0f - wx, iy = 1.0f - wy, iz = 1.0f - wz;

    const int bx = (int)fx, by = (int)fy, bz = (int)fz;

    // Per-axis partial hashes (x-prime is 1: no multiply).
    const uint32_t hx0 = (uint32_t)bx;
    const uint32_t hx1 = (uint32_t)(bx + 1);
    const uint32_t hy0 = (uint32_t)by * P1;
    const uint32_t hy1 = (uint32_t)(by + 1) * P1;
    const uint32_t hz0 = (uint32_t)bz * P2;
    const uint32_t hz1 = (uint32_t)(bz + 1) * P2;

    // Corner order c = i*4 + j*2 + k (matches reference OFFSETS).
    // 8 independent global_load_b64 gathers, default RT hint (L2-resident).
    const F2 e0 = tbl[(hx0 ^ hy0 ^ hz0) & msk];
    const F2 e1 = tbl[(hx0 ^ hy0 ^ hz1) & msk];
    const F2 e2 = tbl[(hx0 ^ hy1 ^ hz0) & msk];
    const F2 e3 = tbl[(hx0 ^ hy1 ^ hz1) & msk];
    const F2 e4 = tbl[(hx1 ^ hy0 ^ hz0) & msk];
    const F2 e5 = tbl[(hx1 ^ hy0 ^ hz1) & msk];
    const F2 e6 = tbl[(hx1 ^ hy1 ^ hz0) & msk];
    const F2 e7 = tbl[(hx1 ^ hy1 ^ hz1) & msk];

    // Trilinear blend, same association order as the reference.
    const float c00a = e0.a * ix + e4.a * wx, c00b = e0.b * ix + e4.b * wx;
    const float c01a = e1.a * ix + e5.a * wx, c01b = e1.b * ix + e5.b * wx;
    const float c10a = e2.a * ix + e6.a * wx, c10b = e2.b * ix + e6.b * wx;
    const float c11a = e3.a * ix + e7.a * wx, c11b = e3.b * ix + e7.b * wx;

    const float c0a = c00a * iy + c10a * wy, c0b = c00b * iy + c10b * wy;
    const float c1a = c01a * iy + c11a * wy, c1b = c01b * iy + c11b * wy;

    ra = c0a * iz + c1a * wz;
    rb = c0b * iz + c1b * wz;
}

__global__ __launch_bounds__(256) void
hashenc_kernel(const float* __restrict__ x,
               const float* __restrict__ tables,
               float* __restrict__ out,
               int n)
{
    const int gid = blockIdx.x * blockDim.x + threadIdx.x;
    if (gid >= n) return;

    // x is a read-once 12 MB stream: NT loads keep L2 for the tables.
    const float* xp = x + (size_t)gid * 3;
    const float px = __builtin_nontemporal_load(xp + 0);
    const float py = __builtin_nontemporal_load(xp + 1);
    const float pz = __builtin_nontemporal_load(xp + 2);
    const float pxp = px + 1.0f;
    const float pyp = py + 1.0f;
    const float pzp = pz + 1.0f;

    float r[L_LEVELS * F_FEAT];

    // Level 0: res=16 -> sc=2^3, gr=2^-3; exact doubling/halving per level.
    // Full unroll: sc/gr fold to literals, r[] indices become constants.
    float sc = 8.0f;
    float gr = 0.125f;
    const F2* __restrict__ tbl = (const F2*)tables;

#pragma unroll
    for (int l = 0; l < L_LEVELS; ++l) {
        level_interp(tbl, px, py, pz, pxp, pyp, pzp, sc, gr,
                     r[2 * l + 0], r[2 * l + 1]);
        tbl += T_SIZE;
        sc *= 2.0f;
        gr *= 0.5f;
    }

    // Write the whole 128 B row with 8 consecutive NT b128 stores: the line
    // is fully written within one clause -> single full-line NT writeback,
    // and the 128 MB output stream never evicts table lines from L2.
    v4f* __restrict__ ov = (v4f*)(out + (size_t)gid * (L_LEVELS * F_FEAT));
#pragma unroll
    for (int i = 0; i < 8; ++i) {
        v4f v;
        v.x = r[4 * i + 0];
        v.y = r[4 * i + 1];
        v.z = r[4 * i + 2];
        v.w = r[4 * i + 3];
        __builtin_nontemporal_store(v, ov + i);
    }
}

extern "C" void kernel_launch(void* const* d_in, const int* in_sizes, int n_in,
                              void* d_out, int out_size, void* d_ws, size_t ws_size,
                              hipStream_t stream)
{
    const float* x      = (const float*)d_in[0];   // (N, 3) float32
    const float* tables = (const float*)d_in[1];   // (L, T, F) float32
    float* out          = (float*)d_out;           // (N, L*F) float32

    const int n = in_sizes[0] / 3;
    const int block = 256;
    const int grid = (n + block - 1) / block;

    hashenc_kernel<<<grid, block, 0, stream>>>(x, tables, out, n);
}